// TrajScoreSelection_79697413145185
// MI455X (gfx1250) — compile-verified
//
#include <hip/hip_runtime.h>
#include <math.h>

#define B_    1024
#define M_    50
#define C_    2048
#define H_    30
#define HID_  512
#define K_    6
#define TW_   60      // H*2
#define TEMPER_ 0.01f
#define THRESH_ 0.1f
#define EPS_    1e-5f

typedef float v2f __attribute__((ext_vector_type(2)));
typedef float v4f __attribute__((ext_vector_type(4)));
typedef float v8f __attribute__((ext_vector_type(8)));
typedef int   v4i __attribute__((vector_size(16)));   // matches builtin param type

#define AS1 __attribute__((address_space(1)))
#define AS3 __attribute__((address_space(3)))

#if defined(__has_builtin)
#  if __has_builtin(__builtin_amdgcn_global_load_async_to_lds_b128)
#    define HAVE_ASYNC_COPY 1
#  endif
#endif
#ifndef HAVE_ASYNC_COPY
#  define HAVE_ASYNC_COPY 0
#endif

// 16-byte global -> LDS copy: async (gfx1250 GLOBAL_LOAD_ASYNC_TO_LDS_B128,
// tracked by ASYNCcnt) when available, else regular load+store.
__device__ inline void copy16_g2l(const float* g, float* l) {
#if HAVE_ASYNC_COPY
  __builtin_amdgcn_global_load_async_to_lds_b128((AS1 v4i*)g, (AS3 v4i*)l, 0, 0);
#else
  *(v4f*)l = *(const v4f*)g;
#endif
}

__device__ inline void wait_async_copies() {
#if HAVE_ASYNC_COPY
#  if __has_builtin(__builtin_amdgcn_s_wait_asynccnt)
  __builtin_amdgcn_s_wait_asynccnt(0);
#  else
  asm volatile("s_wait_asynccnt 0x0" ::: "memory");
#  endif
#endif
}

// ---------------------------------------------------------------------------
// Kernel 1: base = feat_in @ W1[0:C, :]   (B x HID f32 GEMM via WMMA 16x16x4)
// grid: (B/16, HID/128), block 256 (8 waves, each wave one 16x16 tile)
// ---------------------------------------------------------------------------
__global__ __launch_bounds__(256)
void k1_base_gemm(const float* __restrict__ feat, const float* __restrict__ W1,
                  float* __restrict__ base) {
  const int tid   = threadIdx.x;
  const int lane  = tid & 31;
  const int wave  = tid >> 5;
  const int row0  = blockIdx.x * 16;
  const int col0  = blockIdx.y * 128 + wave * 16;
  const int halfk = lane >> 4;
  const int l15   = lane & 15;
  const int arow  = row0 + l15;
  const int bcol  = col0 + l15;

  v8f acc = {0.f, 0.f, 0.f, 0.f, 0.f, 0.f, 0.f, 0.f};
  for (int k = 0; k < C_; k += 4) {
    const int kk = k + 2 * halfk;
    v2f a = *(const v2f*)(feat + (size_t)arow * C_ + kk);   // A: 16x4 frag
    v2f b;
    b.x = W1[(size_t)kk * HID_ + bcol];                      // B: 4x16 frag
    b.y = W1[(size_t)(kk + 1) * HID_ + bcol];
    acc = __builtin_amdgcn_wmma_f32_16x16x4_f32(false, a, false, b,
                                                (short)0, acc, false, false);
  }
#pragma unroll
  for (int v = 0; v < 8; ++v) {
    const int m = v + 8 * halfk;
    base[(size_t)(row0 + m) * HID_ + bcol] = acc[v];
  }
}

// per-row LayerNorm + ReLU over HID_ values; wave handles rows 2w, 2w+1
__device__ inline void ln_relu_rows(float* buf, const float* __restrict__ g,
                                    const float* __restrict__ be,
                                    int wave, int lane) {
#pragma unroll
  for (int rr = 0; rr < 2; ++rr) {
    const int r = wave * 2 + rr;
    float s = 0.f, s2 = 0.f;
    for (int j = lane; j < HID_; j += 32) {
      const float v = buf[r * HID_ + j];
      s += v; s2 += v * v;
    }
#pragma unroll
    for (int off = 16; off > 0; off >>= 1) {
      s  += __shfl_xor(s,  off, 32);
      s2 += __shfl_xor(s2, off, 32);
    }
    const float mu  = s * (1.f / HID_);
    const float var = s2 * (1.f / HID_) - mu * mu;
    const float rs  = rsqrtf(var + EPS_);
    for (int j = lane; j < HID_; j += 32) {
      const float v = (buf[r * HID_ + j] - mu) * rs * g[j] + be[j];
      buf[r * HID_ + j] = v > 0.f ? v : 0.f;
    }
  }
}

// ---------------------------------------------------------------------------
// Kernel 2: fused  h1 = relu(LN(base + traj@W1t + b1)),
//                  h2 = relu(LN(h1 @ W2 + b2))  [LDS-staged WMMA],
//                  logits = h2 @ W3 + b3
// grid: (B*M/16) blocks of 256 threads; each block = 16 rows.
// ---------------------------------------------------------------------------
#define PSTRIDE_ (HID_ + 4)   // padded LDS row stride for the W2 K-panel

__global__ __launch_bounds__(256)
void k2_fused(const float* __restrict__ traj, const float* __restrict__ base,
              const float* __restrict__ W1,  const float* __restrict__ b1,
              const float* __restrict__ g1,  const float* __restrict__ be1,
              const float* __restrict__ W2,  const float* __restrict__ b2,
              const float* __restrict__ g2,  const float* __restrict__ be2,
              const float* __restrict__ W3,  const float* __restrict__ b3,
              float* __restrict__ logits) {
  __shared__ __align__(16) float sTraj[16 * TW_];
  __shared__ __align__(16) float sH1[16 * HID_];
  __shared__ __align__(16) float sH2[16 * HID_];
  __shared__ __align__(16) float sW2[16 * PSTRIDE_];   // 16-row K-panel of W2

  const int tid  = threadIdx.x;
  const int lane = tid & 31;
  const int wave = tid >> 5;
  const int row0 = blockIdx.x * 16;          // global row in [0, B*M)

  for (int i = tid; i < 16 * TW_; i += 256) {
    const int r = i / TW_, c = i % TW_;
    sTraj[i] = traj[(size_t)(row0 + r) * TW_ + c];
  }
  __syncthreads();

  // --- step 1: pre-LN layer1: base[b] + traj @ W1[C:, :] + b1 ---
  for (int idx = tid; idx < 16 * HID_; idx += 256) {
    const int r = idx >> 9, j = idx & (HID_ - 1);
    const int b = (row0 + r) / M_;
    float acc = base[(size_t)b * HID_ + j] + b1[j];
    const float* w = W1 + (size_t)C_ * HID_ + j;
    const float* t = sTraj + r * TW_;
#pragma unroll 4
    for (int q = 0; q < TW_; ++q) acc += t[q] * w[(size_t)q * HID_];
    sH1[idx] = acc;
  }
  __syncthreads();
  ln_relu_rows(sH1, g1, be1, wave, lane);
  __syncthreads();

  // --- step 2: h1tile(16x512) @ W2(512x512), K-panel staged in LDS ---
  const int halfk = lane >> 4;
  const int l15   = lane & 15;
  const int wcol  = wave * 64;               // this wave's 64-column panel
  v8f acc[4] = {};

  for (int kc = 0; kc < HID_; kc += 16) {
    // cooperative 16x512 panel copy: 2048 x 16B, 8 per thread, coalesced
#pragma unroll
    for (int p = 0; p < 8; ++p) {
      const int v    = tid + p * 256;        // vec4 index within panel
      const int krow = v >> 7;               // 128 vec4 per panel row
      const int c4   = (v & 127) << 2;
      copy16_g2l(W2 + (size_t)(kc + krow) * HID_ + c4,
                 sW2 + krow * PSTRIDE_ + c4);
    }
    wait_async_copies();
    __syncthreads();

#pragma unroll
    for (int k4 = 0; k4 < 16; k4 += 4) {
      const int kk = k4 + 2 * halfk;
      v2f a = *(const v2f*)(sH1 + l15 * HID_ + kc + kk);
#pragma unroll
      for (int t = 0; t < 4; ++t) {
        v2f b;
        b.x = sW2[kk * PSTRIDE_ + wcol + t * 16 + l15];
        b.y = sW2[(kk + 1) * PSTRIDE_ + wcol + t * 16 + l15];
        acc[t] = __builtin_amdgcn_wmma_f32_16x16x4_f32(false, a, false, b,
                                                       (short)0, acc[t],
                                                       false, false);
      }
    }
    __syncthreads();   // panel reused next chunk
  }

  // write accumulators (+b2) into sH2
#pragma unroll
  for (int t = 0; t < 4; ++t) {
#pragma unroll
    for (int v = 0; v < 8; ++v) {
      const int m = v + 8 * halfk;
      const int n = wcol + t * 16 + l15;
      sH2[m * HID_ + n] = acc[t][v] + b2[n];
    }
  }
  __syncthreads();
  ln_relu_rows(sH2, g2, be2, wave, lane);
  __syncthreads();

  // --- step 3: logits = h2 @ W3 + b3 ---
#pragma unroll
  for (int rr = 0; rr < 2; ++rr) {
    const int r = wave * 2 + rr;
    float s = 0.f;
    for (int j = lane; j < HID_; j += 32) s += sH2[r * HID_ + j] * W3[j];
#pragma unroll
    for (int off = 16; off > 0; off >>= 1) s += __shfl_xor(s, off, 32);
    if (lane == 0) logits[row0 + r] = s + b3[0];
  }
}

// ---------------------------------------------------------------------------
// Kernel 3: per-batch softmax, loss partial, stable argsort(-score),
//           gathers, serial NMS selection.  grid: B blocks of 64 threads.
// ---------------------------------------------------------------------------
__global__ __launch_bounds__(64)
void k3_select(const float* __restrict__ traj, const float* __restrict__ gt,
               const float* __restrict__ targets, const float* __restrict__ logits,
               float* __restrict__ outSel, float* __restrict__ outTgt,
               float* __restrict__ lossPart) {
  __shared__ float lg[M_], sgl[M_], sp[M_], sg[M_];
  __shared__ int   order[M_];
  __shared__ float sTrajS[M_ * TW_];
  __shared__ int   selIdx[K_];
  __shared__ int   cntSh;

  const int b = blockIdx.x;
  const int tid = threadIdx.x;

  if (tid < M_) {
    lg[tid] = logits[b * M_ + tid];
    const float* tr = traj + (size_t)(b * M_ + tid) * TW_;
    const float* g  = gt + (size_t)b * TW_;
    float dmax = -1e30f;
    for (int h = 0; h < H_; ++h) {
      const float dx = tr[2 * h] - g[2 * h];
      const float dy = tr[2 * h + 1] - g[2 * h + 1];
      dmax = fmaxf(dmax, dx * dx + dy * dy);
    }
    sgl[tid] = -dmax / TEMPER_;
  }
  __syncthreads();

  if (tid < M_) {
    float mx = -1e30f;
    for (int j = 0; j < M_; ++j) mx = fmaxf(mx, lg[j]);
    float sum = 0.f;
    for (int j = 0; j < M_; ++j) sum += expf(lg[j] - mx);
    sp[tid] = expf(lg[tid] - mx) / sum;

    float mx2 = -1e30f;
    for (int j = 0; j < M_; ++j) mx2 = fmaxf(mx2, sgl[j]);
    float sum2 = 0.f;
    for (int j = 0; j < M_; ++j) sum2 += expf(sgl[j] - mx2);
    sg[tid] = expf(sgl[tid] - mx2) / sum2;
  }
  __syncthreads();

  if (tid == 0) {
    float L = 0.f;
    for (int j = 0; j < M_; ++j) { const float d = sp[j] - sg[j]; L += d * d; }
    lossPart[b] = L;
  }
  // stable rank of -score (ties -> lower index first)
  if (tid < M_) {
    const float s = sp[tid];
    int r = 0;
    for (int j = 0; j < M_; ++j)
      r += (sp[j] > s) || (sp[j] == s && j < tid);
    order[r] = tid;
  }
  __syncthreads();

  if (tid < 2 * M_) {
    const int r = tid >> 1, c = tid & 1;
    outTgt[(size_t)(b * M_ + r) * 2 + c] =
        targets[(size_t)(b * M_ + order[r]) * 2 + c];
  }
  for (int i = tid; i < M_ * TW_; i += 64) {
    const int r = i / TW_, c = i % TW_;
    sTrajS[i] = traj[(size_t)(b * M_ + order[r]) * TW_ + c];
  }
  __syncthreads();

  if (tid == 0) {
    int sel[K_];
    int cnt = 1;
    sel[0] = 0;
    for (int i = 1; i < M_; ++i) {
      bool close = false;
      for (int j = 0; j < cnt && !close; ++j) {
        float dmax = -1e30f;
        for (int h = 0; h < H_; ++h) {
          const float dx = sTrajS[sel[j] * TW_ + 2 * h]     - sTrajS[i * TW_ + 2 * h];
          const float dy = sTrajS[sel[j] * TW_ + 2 * h + 1] - sTrajS[i * TW_ + 2 * h + 1];
          dmax = fmaxf(dmax, dx * dx + dy * dy);
        }
        if (dmax < THRESH_) close = true;
      }
      if (!close && cnt < K_) { sel[cnt] = i; ++cnt; }
    }
    for (int j = 0; j < K_; ++j) selIdx[j] = (j < cnt) ? sel[j] : 0;
    cntSh = cnt;
  }
  __syncthreads();

  for (int i = tid; i < K_ * TW_; i += 64) {
    const int r = i / TW_, c = i % TW_;
    outSel[(size_t)(b * K_ + r) * TW_ + c] =
        (r < cntSh) ? sTrajS[selIdx[r] * TW_ + c] : 0.f;
  }
}

// ---------------------------------------------------------------------------
// Kernel 4: reduce per-batch loss partials -> scalar loss
// ---------------------------------------------------------------------------
__global__ __launch_bounds__(256)
void k4_loss(const float* __restrict__ lossPart, float* __restrict__ outLoss) {
  __shared__ float red[256];
  float s = 0.f;
  for (int i = threadIdx.x; i < B_; i += 256) s += lossPart[i];
  red[threadIdx.x] = s;
  __syncthreads();
  for (int o = 128; o > 0; o >>= 1) {
    if (threadIdx.x < o) red[threadIdx.x] += red[threadIdx.x + o];
    __syncthreads();
  }
  if (threadIdx.x == 0) outLoss[0] = red[0] / (float)(B_ * M_);
}

// ---------------------------------------------------------------------------
extern "C" void kernel_launch(void* const* d_in, const int* in_sizes, int n_in,
                              void* d_out, int out_size, void* d_ws, size_t ws_size,
                              hipStream_t stream) {
  (void)in_sizes; (void)n_in; (void)out_size; (void)ws_size;
  const float* feat    = (const float*)d_in[0];
  const float* traj    = (const float*)d_in[1];
  const float* gt      = (const float*)d_in[2];
  const float* targets = (const float*)d_in[3];
  const float* W1  = (const float*)d_in[4];
  const float* b1  = (const float*)d_in[5];
  const float* g1  = (const float*)d_in[6];
  const float* be1 = (const float*)d_in[7];
  const float* W2  = (const float*)d_in[8];
  const float* b2  = (const float*)d_in[9];
  const float* g2  = (const float*)d_in[10];
  const float* be2 = (const float*)d_in[11];
  const float* W3  = (const float*)d_in[12];
  const float* b3  = (const float*)d_in[13];

  float* out     = (float*)d_out;
  float* outSel  = out;                               // B*K*60
  float* outTgt  = out + (size_t)B_ * K_ * TW_;       // B*M*2
  float* outLoss = outTgt + (size_t)B_ * M_ * 2;      // 1

  float* ws       = (float*)d_ws;
  float* wsBase   = ws;                               // B*HID
  float* wsLogits = wsBase + (size_t)B_ * HID_;       // B*M
  float* wsLossP  = wsLogits + (size_t)B_ * M_;       // B

  k1_base_gemm<<<dim3(B_ / 16, HID_ / 128), 256, 0, stream>>>(feat, W1, wsBase);
  k2_fused<<<dim3((B_ * M_) / 16), 256, 0, stream>>>(
      traj, wsBase, W1, b1, g1, be1, W2, b2, g2, be2, W3, b3, wsLogits);
  k3_select<<<dim3(B_), 64, 0, stream>>>(traj, gt, targets, wsLogits,
                                         outSel, outTgt, wsLossP);
  k4_loss<<<1, 256, 0, stream>>>(wsLossP, outLoss);
}